// BTVLoss_24404004176391
// MI455X (gfx1250) — compile-verified
//
#include <hip/hip_runtime.h>

// BTV loss, x = (8,3,1024,1024) f32.
// VALU-bound problem (48x arithmetic amplification; 100MB footprint fits L2).
// Optimizations:
//  - +/- shift-pair symmetry: compute 24 shifts, scale by 2 (halves all work).
//  - LDS tile (16x64 interior + 3 bottom/left/right halo) so every neighbor is a
//    ds_load_b32 with an immediate offset: zero per-neighbor address VALU.
//  - 7x7 register window per thread, CSE'd across its 4 output rows.
//  - raw v_sqrt_f32 via __builtin_amdgcn_sqrtf (1 instr, ~1ulp, fine for a loss).
//  - wave reduction via V_WMMA_F32_16X16X4_F32 (ones x partials), CDNA5 matrix path.
//  - deterministic 2-pass reduction through d_ws (graph-replay safe, no atomics).

#define EPS 1e-6f
#define TW 64
#define TH 16
#define LDSW 70      // TW + 3 left + 3 right halo
#define LDSH 19      // TH + 3 bottom halo (no top halo needed with +k,+l offsets)
#define HMASK 1023

typedef float v2f __attribute__((ext_vector_type(2)));
typedef float v8f __attribute__((ext_vector_type(8)));

// Reduce 32 lane values to the wave total using one WMMA f32 16x16x4.
// A = ones(16x4), B holds the 32 accs (+32 zeros) in its 64 slots (any bijective
// lane layout works): D[m][n] = sum_k B[k][n], so summing row 0 (lanes 0..15 of
// d[0]) over n gives the total of all B entries. 4 xor-shuffles finish.
__device__ __forceinline__ float wave_reduce_wmma(float acc) {
  v2f a = {1.0f, 1.0f};
  v2f b = {acc, 0.0f};
  v8f c = {};
  c = __builtin_amdgcn_wmma_f32_16x16x4_f32(
      /*neg_a=*/false, a, /*neg_b=*/false, b,
      /*c_mod=*/(short)0, c, /*reuse_a=*/false, /*reuse_b=*/false);
  float v = c[0];
  v += __shfl_xor(v, 8, 32);
  v += __shfl_xor(v, 4, 32);
  v += __shfl_xor(v, 2, 32);
  v += __shfl_xor(v, 1, 32);
  return v;  // every lane holds the wave total
}

__global__ __launch_bounds__(256) void btv_stage1(const float* __restrict__ x,
                                                  float* __restrict__ partial) {
  __shared__ float tile[LDSH * LDSW];
  __shared__ float wsum[8];

  const int bid = blockIdx.x;
  const int plane = bid >> 10;            // 24 planes, 1024 tiles each
  const int t = bid & 1023;
  const int tileC0 = (t & 15) << 6;       // 16 tiles across, 64 cols each
  const int tileR0 = (t >> 4) << 4;       // 64 tiles down, 16 rows each
  const float* p = x + (size_t)plane * (1024u * 1024u);

  // Load 19x70 tile (wrap-around roll handled here, once, with &1023).
  for (int idx = threadIdx.x; idx < LDSH * LDSW; idx += 256) {
    const int lr = idx / LDSW;
    const int lc = idx - lr * LDSW;
    const int gr = (tileR0 + lr) & HMASK;
    const int gc = (tileC0 - 3 + lc) & HMASK;
    tile[idx] = p[(gr << 10) + gc];
  }
  // CDNA5 global_prefetch_b8: hint the row block below this tile.
  __builtin_prefetch(p + (((tileR0 + TH) & HMASK) << 10) + tileC0, 0, 1);
  __syncthreads();

  const int tx = threadIdx.x & 63;        // column 0..63
  const int tq = threadIdx.x >> 6;        // row group 0..3 -> rows 4*tq..4*tq+3
  const float* base = &tile[(tq * 4) * LDSW + tx];

  // 7x7 register window: all ds_load_b32 with non-negative immediate offsets.
  float win[7][7];
#pragma unroll
  for (int i = 0; i < 7; ++i)
#pragma unroll
    for (int j = 0; j < 7; ++j)
      win[i][j] = base[i * LDSW + j];

  float acc = 0.0f;
#pragma unroll
  for (int rr = 0; rr < 4; ++rr) {
    const float c0 = win[rr][3];
    // k = 0, l = 1..3 (one of each +/- pair)
#pragma unroll
    for (int l = 1; l <= 3; ++l) {
      const float d = c0 - win[rr][3 + l];
      acc += __builtin_amdgcn_sqrtf(__builtin_fmaf(d, d, EPS));
    }
    // k = 1..3, l = -3..3
#pragma unroll
    for (int k = 1; k <= 3; ++k)
#pragma unroll
      for (int l = -3; l <= 3; ++l) {
        const float d = c0 - win[rr + k][3 + l];
        acc += __builtin_amdgcn_sqrtf(__builtin_fmaf(d, d, EPS));
      }
  }

  // Block reduction: WMMA per-wave total, then 8 wave sums via LDS.
  const float wv = wave_reduce_wmma(acc);
  const int lane = threadIdx.x & 31;
  const int wid = threadIdx.x >> 5;
  if (lane == 0) wsum[wid] = wv;
  __syncthreads();
  if (threadIdx.x == 0) {
    float s = 0.0f;
#pragma unroll
    for (int i = 0; i < 8; ++i) s += wsum[i];
    partial[bid] = s;
  }
}

__global__ __launch_bounds__(256) void btv_stage2(const float* __restrict__ partial,
                                                  int n, float* __restrict__ out,
                                                  float scale) {
  __shared__ float wsum[8];
  float acc = 0.0f;
  for (int i = threadIdx.x; i < n; i += 256) acc += partial[i];
  const float wv = wave_reduce_wmma(acc);
  const int lane = threadIdx.x & 31;
  const int wid = threadIdx.x >> 5;
  if (lane == 0) wsum[wid] = wv;
  __syncthreads();
  if (threadIdx.x == 0) {
    float s = 0.0f;
#pragma unroll
    for (int i = 0; i < 8; ++i) s += wsum[i];
    out[0] = s * scale;
  }
}

extern "C" void kernel_launch(void* const* d_in, const int* in_sizes, int n_in,
                              void* d_out, int out_size, void* d_ws, size_t ws_size,
                              hipStream_t stream) {
  (void)in_sizes; (void)n_in; (void)out_size; (void)ws_size;
  const float* x = (const float*)d_in[0];
  float* out = (float*)d_out;
  float* partial = (float*)d_ws;   // 24576 floats = 96 KB scratch

  const int nblocks = 24 * 1024;   // 24 planes x (16 x 64) tiles
  btv_stage1<<<nblocks, 256, 0, stream>>>(x, partial);
  // x2 for the +/- shift-pair symmetry, x WEIGHT / (b*c*h*w)
  const float scale = 2.0f * 0.1f / (float)(8 * 3 * 1024 * 1024);
  btv_stage2<<<1, 256, 0, stream>>>(partial, nblocks, out, scale);
}